// Set2Vec_11819749999299
// MI455X (gfx1250) — compile-verified
//
#include <hip/hip_runtime.h>
#include <math.h>

#define B_      512
#define N_      512
#define NF      128
#define TWO_NF  256
#define STEPS   100
#define BIG_NEG -100000000.0f

typedef __attribute__((ext_vector_type(2))) float v2f;
typedef __attribute__((ext_vector_type(4))) float v4f;
typedef __attribute__((ext_vector_type(8))) float v8f;

__device__ __forceinline__ float sigmoidf_(float x) { return 1.0f / (1.0f + __expf(-x)); }

__device__ __forceinline__ float wave_sum32(float v) {
#pragma unroll
    for (int off = 16; off > 0; off >>= 1) v += __shfl_xor(v, off, 32);
    return v;
}

// ---------------- utility kernels ----------------
__global__ void zero_kernel(float* p, int n) {
    int i = blockIdx.x * blockDim.x + threadIdx.x;
    if (i < n) p[i] = 0.0f;
}

__global__ void copy_kernel(const float* __restrict__ src, float* __restrict__ dst, int n) {
    int i = blockIdx.x * blockDim.x + threadIdx.x;
    if (i < n) dst[i] = src[i];
}

// Pack gate weights K-major: W4T[col(512)][k(256)], col = g*128+j.
// Also pack wqT[n(128)][k(128)] and b4[512].
__global__ void pack_weights(const float* __restrict__ wi, const float* __restrict__ wf,
                             const float* __restrict__ wg, const float* __restrict__ wo,
                             const float* __restrict__ bi, const float* __restrict__ bf,
                             const float* __restrict__ bg, const float* __restrict__ bo,
                             const float* __restrict__ wq,
                             float* __restrict__ W4T, float* __restrict__ wqT,
                             float* __restrict__ b4) {
    int idx = blockIdx.x * blockDim.x + threadIdx.x;   // 131072 threads
    if (idx >= 512 * TWO_NF) return;
    int col = idx >> 8;          // 0..511
    int k   = idx & 255;         // 0..255
    int g = col >> 7;
    int j = col & 127;
    const float* w = (g == 0) ? wi : (g == 1) ? wf : (g == 2) ? wg : wo;
    W4T[idx] = w[k * NF + j];    // = B[k, col]
    if (idx < NF * NF) {
        int n = idx >> 7, kk = idx & 127;
        wqT[idx] = wq[kk * NF + n];
    }
    if (idx < 4 * NF) {
        int gg = idx >> 7, jj = idx & 127;
        const float* bb = (gg == 0) ? bi : (gg == 1) ? bf : (gg == 2) ? bg : bo;
        b4[idx] = bb[jj];
    }
}

// ---------------- WMMA f32 GEMM with K-major B: C[M,N] = A[M,K] * BT[N,K]^T
// One wave computes a 16x32 tile (two 16x16 accumulators sharing the A fragment).
// Per K-step: 3x global_load_b64 + 2x v_wmma_f32_16x16x4_f32.
__global__ void wmma_gemm_f32_bt(const float* __restrict__ A, int lda,
                                 const float* __restrict__ BT, int ldb,
                                 float* __restrict__ C, int ldc,
                                 int tiles_n32, int K) {
    int lane = threadIdx.x & 31;
    int wave = threadIdx.x >> 5;
    int tile = blockIdx.x * 4 + wave;
    int tm = tile / tiles_n32;
    int tn = tile % tiles_n32;
    int m0 = tm * 16, n0 = tn * 32;
    int half = lane >> 4;   // lanes 16-31 hold K+2/K+3
    int r = lane & 15;      // A row / B col within tile
    v8f acc0 = {}, acc1 = {};
    const float* arow  = A  + (m0 + r) * lda + half * 2;
    const float* brow0 = BT + (n0 + r) * ldb + half * 2;
    const float* brow1 = BT + (n0 + 16 + r) * ldb + half * 2;
    for (int k = 0; k < K; k += 4) {
        v2f a  = *(const v2f*)(arow + k);    // A[m, k+2h], A[m, k+2h+1]
        v2f b0 = *(const v2f*)(brow0 + k);   // B[k+2h, n], B[k+2h+1, n]
        v2f b1 = *(const v2f*)(brow1 + k);
        acc0 = __builtin_amdgcn_wmma_f32_16x16x4_f32(
            false, a, false, b0, (short)0, acc0, false, false);
        acc1 = __builtin_amdgcn_wmma_f32_16x16x4_f32(
            false, a, false, b1, (short)0, acc1, false, false);
    }
    // D layout: VGPR i -> (m0+i, n0+lane) lanes0-15 ; (m0+8+i, .) lanes16-31
    float* crow0 = C + (m0 + half * 8) * ldc + n0 + r;
    float* crow1 = crow0 + 16;
#pragma unroll
    for (int i = 0; i < 8; ++i) {
        crow0[i * ldc] = acc0[i];
        crow1[i * ldc] = acc1[i];
    }
}

// ---------------- LSTM cell elementwise ----------------
__global__ void lstm_cell(const float* __restrict__ pre, const float* __restrict__ b4,
                          float* __restrict__ c, float* __restrict__ h) {
    int idx = blockIdx.x * blockDim.x + threadIdx.x;   // 512*128
    if (idx >= B_ * NF) return;
    int b = idx >> 7, f = idx & 127;
    const float* row = pre + b * 512;
    float ig = sigmoidf_(row[f]       + b4[f]);
    float fg = sigmoidf_(row[128 + f] + b4[128 + f]);
    float gg = tanhf   (row[256 + f]  + b4[256 + f]);
    float og = sigmoidf_(row[384 + f] + b4[384 + f]);
    float cn = fg * c[idx] + ig * gg;
    c[idx] = cn;
    h[idx] = og * tanhf(cn);
}

// ---------------- attention energies + per-block max ----------------
// One wave per (b,n) pair: 128-wide dot of we * tanh(q+x) via float4 + shuffle reduce.
// 8 waves/block, 8 pairs/wave -> 64 pairs per block.
__global__ void energies_kernel(const float* __restrict__ x, const float* __restrict__ q,
                                const float* __restrict__ we, const unsigned char* __restrict__ mask,
                                float* __restrict__ e, float* __restrict__ partmax) {
    int lane = threadIdx.x & 31;
    int wave = threadIdx.x >> 5;
    int base = (blockIdx.x * 8 + wave) * 8;
    v4f wev = *(const v4f*)(we + lane * 4);
    float wmax = -3.0e38f;
#pragma unroll
    for (int i = 0; i < 8; ++i) {
        int p = base + i;
        int b = p >> 9;                 // N = 512
        v4f xv = *(const v4f*)(x + (size_t)p * NF + lane * 4);
        v4f qv = *(const v4f*)(q + b * NF + lane * 4);
        float s = wev.x * tanhf(qv.x + xv.x) + wev.y * tanhf(qv.y + xv.y) +
                  wev.z * tanhf(qv.z + xv.z) + wev.w * tanhf(qv.w + xv.w);
        s = wave_sum32(s);
        s += (1.0f - (float)mask[p]) * BIG_NEG;
        if (lane == 0) e[p] = s;
        wmax = fmaxf(wmax, s);
    }
    __shared__ float sm[8];
    if (lane == 0) sm[wave] = wmax;
    __syncthreads();
    if (threadIdx.x == 0) {
        float m = sm[0];
#pragma unroll
        for (int i = 1; i < 8; ++i) m = fmaxf(m, sm[i]);
        partmax[blockIdx.x] = m;
    }
}

// reduce 4096 partial maxima -> scal[0]
__global__ void reduce_max_kernel(const float* __restrict__ part, float* __restrict__ scal) {
    __shared__ float sm[256];
    float m = -3.0e38f;
    for (int i = threadIdx.x; i < 4096; i += 256) m = fmaxf(m, part[i]);
    sm[threadIdx.x] = m;
    __syncthreads();
    for (int s = 128; s > 0; s >>= 1) {
        if (threadIdx.x < s) sm[threadIdx.x] = fmaxf(sm[threadIdx.x], sm[threadIdx.x + s]);
        __syncthreads();
    }
    if (threadIdx.x == 0) scal[0] = sm[0];
}

// p = exp(e - gmax) in place + per-block partial sum
__global__ void expsum_kernel(float* __restrict__ e, const float* __restrict__ scal,
                              float* __restrict__ partsum) {
    int i = blockIdx.x * 256 + threadIdx.x;
    float gmax = scal[0];
    float v = __expf(e[i] - gmax);
    e[i] = v;
    v = wave_sum32(v);
    __shared__ float sm[8];
    if ((threadIdx.x & 31) == 0) sm[threadIdx.x >> 5] = v;
    __syncthreads();
    if (threadIdx.x == 0) {
        float s = sm[0];
#pragma unroll
        for (int k = 1; k < 8; ++k) s += sm[k];
        partsum[blockIdx.x] = s;
    }
}

// reduce 1024 partial sums -> scal[1]
__global__ void reduce_sum_kernel(const float* __restrict__ part, float* __restrict__ scal) {
    __shared__ float sm[256];
    float s = 0.0f;
    for (int i = threadIdx.x; i < 1024; i += 256) s += part[i];
    sm[threadIdx.x] = s;
    __syncthreads();
    for (int k = 128; k > 0; k >>= 1) {
        if (threadIdx.x < k) sm[threadIdx.x] += sm[threadIdx.x + k];
        __syncthreads();
    }
    if (threadIdx.x == 0) scal[1] = sm[0];
}

// read[b,f] = (1/S) * sum_n p[b,n] * x[b,n,f]; writes m_next = [h | read]
__global__ void read_kernel(const float* __restrict__ x, const float* __restrict__ p,
                            const float* __restrict__ h, const float* __restrict__ scal,
                            float* __restrict__ m) {
    int b = blockIdx.x;         // 512 blocks
    int f = threadIdx.x;        // 128 threads
    __shared__ float pl[N_];
    for (int i = f; i < N_; i += 128) pl[i] = p[b * N_ + i];
    __syncthreads();
    float acc = 0.0f;
    const float* xb = x + (size_t)b * N_ * NF + f;
    for (int n = 0; n < N_; ++n) acc += pl[n] * xb[(size_t)n * NF];
    float inv = 1.0f / scal[1];
    m[b * TWO_NF + f] = h[b * NF + f];
    m[b * TWO_NF + NF + f] = acc * inv;
}

// ---------------- host orchestration ----------------
extern "C" void kernel_launch(void* const* d_in, const int* in_sizes, int n_in,
                              void* d_out, int out_size, void* d_ws, size_t ws_size,
                              hipStream_t stream) {
    const float* x    = (const float*)d_in[0];                 // [512,512,128]
    const unsigned char* mask = (const unsigned char*)d_in[1]; // bool [512,512]
    const float* w_hi = (const float*)d_in[2];
    const float* b_hi = (const float*)d_in[3];
    const float* w_hf = (const float*)d_in[4];
    const float* b_hf = (const float*)d_in[5];
    const float* w_hg = (const float*)d_in[6];
    const float* b_hg = (const float*)d_in[7];
    const float* w_ho = (const float*)d_in[8];
    const float* b_ho = (const float*)d_in[9];
    const float* wq   = (const float*)d_in[10];                // [128,128]
    const float* we   = (const float*)d_in[11];                // [128]
    float* out = (float*)d_out;                                // [512,256]

    float* ws = (float*)d_ws;
    float* W4T     = ws;                 // 512*256 (K-major gate weights)
    float* wqT     = W4T + 131072;       // 128*128 (K-major wq)
    float* b4      = wqT + 16384;        // 512
    float* m       = b4 + 512;           // 512*256
    float* c       = m + 131072;         // 512*128
    float* h       = c + 65536;          // 512*128
    float* pre     = h + 65536;          // 512*512
    float* q       = pre + 262144;       // 512*128
    float* e       = q + 65536;          // 512*512 (exp written in place)
    float* partmax = e + 262144;         // 4096
    float* partsum = partmax + 4096;     // 1024
    float* scal    = partsum + 1024;     // 2

    pack_weights<<<512, 256, 0, stream>>>(w_hi, w_hf, w_hg, w_ho,
                                          b_hi, b_hf, b_hg, b_ho, wq,
                                          W4T, wqT, b4);
    zero_kernel<<<512, 256, 0, stream>>>(m, 131072);
    zero_kernel<<<256, 256, 0, stream>>>(c, 65536);

    for (int t = 0; t < STEPS; ++t) {
        // gates: pre[512,512] = m[512,256] @ W4 ; 32x16 tiles of 16x32 -> 512 tiles / 4 per block
        wmma_gemm_f32_bt<<<128, 128, 0, stream>>>(m, TWO_NF, W4T, TWO_NF, pre, 512, 16, TWO_NF);
        lstm_cell<<<256, 256, 0, stream>>>(pre, b4, c, h);
        // query: q[512,128] = h[512,128] @ wq ; 32x4 tiles -> 128 tiles / 4 per block
        wmma_gemm_f32_bt<<<32, 128, 0, stream>>>(h, NF, wqT, NF, q, NF, 4, NF);
        energies_kernel<<<4096, 256, 0, stream>>>(x, q, we, mask, e, partmax);
        reduce_max_kernel<<<1, 256, 0, stream>>>(partmax, scal);
        expsum_kernel<<<1024, 256, 0, stream>>>(e, scal, partsum);
        reduce_sum_kernel<<<1, 256, 0, stream>>>(partsum, scal);
        read_kernel<<<512, 128, 0, stream>>>(x, e, h, scal, m);
    }
    copy_kernel<<<512, 256, 0, stream>>>(m, out, 131072);
}